// GCN_6107443494988
// MI455X (gfx1250) — compile-verified
//
#include <hip/hip_runtime.h>

typedef float v2f __attribute__((ext_vector_type(2)));
typedef float v8f __attribute__((ext_vector_type(8)));

#define NNODES 65536
#define NEDGES 1048576

// ---------------------------------------------------------------------------
// Hardware f32 atomic add (GLOBAL_ATOMIC_ADD_F32, no-return -> STOREcnt,
// implicit wait at S_ENDPGM). Avoids any CAS-loop fallback.
// ---------------------------------------------------------------------------
__device__ __forceinline__ void gadd_f32(float* p, float v) {
    asm volatile("global_atomic_add_f32 %0, %1, off" : : "v"(p), "v"(v) : "memory");
}

// ---------------------------------------------------------------------------
// S[N,64] = H[N,K] @ W[K,64] using V_WMMA_F32_16X16X4_F32 (fp32 in, fp32 acc).
// Block = 256 threads = 8 waves; each wave computes one 16x16 tile.
// Block covers 32 rows x 64 cols; grid.x = N/32.
// A layout (16x4 f32): lanes 0-15 hold K=k+0/k+1 (vgpr0/1), lanes 16-31 K=k+2/k+3.
// B layout (4x16):     vgpr0 = rows k+0 (lanes<16) / k+2 (lanes>=16), vgpr1 = k+1/k+3.
// D layout: vgpr v -> row (v + 8*laneHalf), col = lane&15.
// ---------------------------------------------------------------------------
__global__ __launch_bounds__(256) void gemm_f32_wmma(const float* __restrict__ H,
                                                     const float* __restrict__ Wm,
                                                     float* __restrict__ S, int K) {
    const int lane = threadIdx.x & 31;
    const int wave = threadIdx.x >> 5;
    const int row0 = blockIdx.x * 32 + (wave >> 2) * 16;
    const int col0 = (wave & 3) * 16;
    const int half = lane >> 4;      // 0: lanes 0-15, 1: lanes 16-31
    const int l    = lane & 15;

    const float* hrow = H + (size_t)(row0 + l) * K + 2 * half;   // A: row row0+l
    const float* wcol = Wm + (size_t)(2 * half) * 64 + col0 + l; // B: col col0+l

    v8f acc = {0.f, 0.f, 0.f, 0.f, 0.f, 0.f, 0.f, 0.f};

    for (int k = 0; k < K; k += 4) {
        v2f a = *(const v2f*)(hrow + k);                 // K = k+2h, k+2h+1
        v2f b;
        b.x = wcol[(size_t)k * 64];                      // row k+2h
        b.y = wcol[(size_t)(k + 1) * 64];                // row k+2h+1
        acc = __builtin_amdgcn_wmma_f32_16x16x4_f32(
            /*neg_a=*/false, a, /*neg_b=*/false, b,
            /*c_mod=*/(short)0, acc, /*reuse_a=*/false, /*reuse_b=*/false);
    }

    float* srow = S + (size_t)(row0 + half * 8) * 64 + col0 + l;
#pragma unroll
    for (int v = 0; v < 8; ++v) srow[(size_t)v * 64] = acc[v];
}

// ---------------------------------------------------------------------------
// Zero fill (float4 stores). count4 * 4 floats total; grid.x = count4/256.
// ---------------------------------------------------------------------------
__global__ __launch_bounds__(256) void fill_zero4(float4* __restrict__ p) {
    p[(size_t)blockIdx.x * 256 + threadIdx.x] = make_float4(0.f, 0.f, 0.f, 0.f);
}

// Zero-pad W7 (64x40) into WP (64x64). 4096 threads.
__global__ __launch_bounds__(256) void pad_w7(const float* __restrict__ W7,
                                              float* __restrict__ WP) {
    int i = blockIdx.x * 256 + threadIdx.x;
    int r = i >> 6, c = i & 63;
    WP[i] = (c < 40) ? W7[r * 40 + c] : 0.f;
}

// ---------------------------------------------------------------------------
// Edge aggregation: G[dst[e], :] += S[src[e], :] * ew[e]
// FO4 float4-groups per edge (16 for 64 feats, 10 for 40). Row stride fixed 64.
// Grid.x = E*FO4/256 (exact: both 16 and 10 divide cleanly).
// ---------------------------------------------------------------------------
template <int FO4>
__global__ __launch_bounds__(256) void edge_aggregate(const float* __restrict__ S,
                                                      const int* __restrict__ src,
                                                      const int* __restrict__ dst,
                                                      const float* __restrict__ ew,
                                                      float* __restrict__ G) {
    const long long tid = (long long)blockIdx.x * 256 + threadIdx.x;
    const int e = (int)(tid / FO4);
    const int q = (int)(tid % FO4);
    const int s = src[e];
    const int d = dst[e];
    const float w = ew[e];
    const float4 m = *(const float4*)(S + (size_t)s * 64 + q * 4);
    float* gp = G + (size_t)d * 64 + q * 4;
    gadd_f32(gp + 0, m.x * w);
    gadd_f32(gp + 1, m.y * w);
    gadd_f32(gp + 2, m.z * w);
    gadd_f32(gp + 3, m.w * w);
}

// ---------------------------------------------------------------------------
// out = relu(G + bias) [+ res].  N*64 floats as N*16 float4; grid.x = N*16/256.
// ---------------------------------------------------------------------------
__global__ __launch_bounds__(256) void bias_relu_res(const float* __restrict__ G,
                                                     const float* __restrict__ bias,
                                                     const float* __restrict__ res,
                                                     float* __restrict__ out,
                                                     int has_res) {
    const size_t i = (size_t)blockIdx.x * 256 + threadIdx.x;   // float4 index
    const int f4 = ((int)i & 15) * 4;
    float4 g  = *(const float4*)(G + i * 4);
    float4 bb = *(const float4*)(bias + f4);
    g.x = fmaxf(g.x + bb.x, 0.f);
    g.y = fmaxf(g.y + bb.y, 0.f);
    g.z = fmaxf(g.z + bb.z, 0.f);
    g.w = fmaxf(g.w + bb.w, 0.f);
    if (has_res) {
        const float4 r = *(const float4*)(res + i * 4);
        g.x += r.x; g.y += r.y; g.z += r.z; g.w += r.w;
    }
    *(float4*)(out + i * 4) = g;
}

// ---------------------------------------------------------------------------
// log_softmax over 40 classes. One wave32 per row; lane holds class `lane`
// and (lane<8) class `32+lane`. Reductions via __shfl_xor (warpSize==32).
// G row stride = 64 (cols 40..63 unused). Grid.x = N/8, block = 256 (8 waves).
// ---------------------------------------------------------------------------
__global__ __launch_bounds__(256) void log_softmax40(const float* __restrict__ G,
                                                     const float* __restrict__ b7,
                                                     float* __restrict__ out) {
    const int lane = threadIdx.x & 31;
    const int wave = threadIdx.x >> 5;
    const int row  = blockIdx.x * 8 + wave;
    const float* g = G + (size_t)row * 64;

    const float v0 = g[lane] + b7[lane];
    const float v1 = (lane < 8) ? (g[32 + lane] + b7[32 + lane]) : -__builtin_inff();

    float m = fmaxf(v0, v1);
#pragma unroll
    for (int off = 16; off; off >>= 1) m = fmaxf(m, __shfl_xor(m, off, 32));

    float s = expf(v0 - m) + ((lane < 8) ? expf(v1 - m) : 0.f);
#pragma unroll
    for (int off = 16; off; off >>= 1) s += __shfl_xor(s, off, 32);

    const float ls = m + logf(s);
    out[(size_t)row * 40 + lane] = v0 - ls;
    if (lane < 8) out[(size_t)row * 40 + 32 + lane] = v1 - ls;
}

// ---------------------------------------------------------------------------
extern "C" void kernel_launch(void* const* d_in, const int* in_sizes, int n_in,
                              void* d_out, int out_size, void* d_ws, size_t ws_size,
                              hipStream_t stream) {
    (void)in_sizes; (void)n_in; (void)out_size; (void)ws_size;

    const float* x   = (const float*)d_in[0];
    const int*   src = (const int*)d_in[1];
    const int*   dst = (const int*)d_in[2];
    const float* ew  = (const float*)d_in[3];
    const float* W[7]; const float* B[7];
    for (int i = 0; i < 7; ++i) {
        W[i] = (const float*)d_in[4 + 2 * i];
        B[i] = (const float*)d_in[5 + 2 * i];
    }
    float* out = (float*)d_out;

    const size_t NH = (size_t)NNODES * 64;
    float* S  = (float*)d_ws;            // support  [N,64]
    float* G  = S  + NH;                 // agg      [N,64]
    float* H0 = G  + NH;                 // hidden ping
    float* H1 = H0 + NH;                 // hidden pong
    float* WP = H1 + NH;                 // padded W7 [64,64]

    const int gGemm = NNODES / 32;           // 2048
    const int gElt  = (NNODES * 16) / 256;   // 4096 (N*64 floats as float4)
    const int gAg16 = (NEDGES * 16) / 256;   // 65536
    const int gAg10 = (NEDGES * 10) / 256;   // 40960
    const int gSm   = NNODES / 8;            // 8192

    pad_w7<<<16, 256, 0, stream>>>(W[6], WP);

    // hidden layer: hout = relu( scatter(gemm(h,W)) + b ) [+ res]
    auto layer = [&](const float* h, int K, const float* Wm, const float* bias,
                     const float* res, float* hout) {
        gemm_f32_wmma<<<gGemm, 256, 0, stream>>>(h, Wm, S, K);
        fill_zero4<<<gElt, 256, 0, stream>>>((float4*)G);
        edge_aggregate<16><<<gAg16, 256, 0, stream>>>(S, src, dst, ew, G);
        bias_relu_res<<<gElt, 256, 0, stream>>>(G, bias, res, hout, res != nullptr);
    };

    layer(x,  128, W[0], B[0], nullptr, H0);  // h1 = relu(gc(x))
    layer(H0,  64, W[1], B[1], nullptr, H1);  // h2 = relu(gc(h1))
    layer(H1,  64, W[2], B[2], H1,      H0);  // h3 = relu(gc(h2)) + h2
    layer(H0,  64, W[3], B[3], H0,      H1);  // h4 = relu(gc(h3)) + h3
    layer(H1,  64, W[4], B[4], H1,      H0);  // h5 = relu(gc(h4)) + h4
    layer(H0,  64, W[5], B[5], nullptr, H1);  // h6 = relu(gc(h5))

    // layer 7: logits = scatter(gemm(h6, W7pad)) ; then + b7 and log_softmax
    gemm_f32_wmma<<<gGemm, 256, 0, stream>>>(H1, WP, S, 64);
    fill_zero4<<<gElt, 256, 0, stream>>>((float4*)G);
    edge_aggregate<10><<<gAg10, 256, 0, stream>>>(S, src, dst, ew, G);
    log_softmax40<<<gSm, 256, 0, stream>>>(G, B[6], out);
}